// LinkPredictionModel_64716567216668
// MI455X (gfx1250) — compile-verified
//
#include <hip/hip_runtime.h>
#include <hip/hip_bf16.h>

typedef float v2f __attribute__((ext_vector_type(2)));
typedef float v8f __attribute__((ext_vector_type(8)));

// ---------------------------------------------------------------------------
// Degree / normalization kernels
// ---------------------------------------------------------------------------
__global__ void deg_init_kernel(float* __restrict__ deg, int n) {
    int i = blockIdx.x * blockDim.x + threadIdx.x;
    if (i < n) deg[i] = 1.0f;   // self-loop contributes 1 to every node's degree
}

__global__ void deg_accum_kernel(const int* __restrict__ dst, float* __restrict__ deg, int E) {
    int e = blockIdx.x * blockDim.x + threadIdx.x;
    if (e < E) atomicAdd(&deg[dst[e]], 1.0f);
}

__global__ void dinv_kernel(const float* __restrict__ deg, float* __restrict__ dinv, int n) {
    int i = blockIdx.x * blockDim.x + threadIdx.x;
    if (i < n) dinv[i] = rsqrtf(deg[i]);   // deg >= 1 always (self-loop)
}

// ---------------------------------------------------------------------------
// f32 WMMA GEMM:  C[M,N] = A[M,K] * B[K,N]   (row-major, no bias)
// Block: 256 threads = 8 waves arranged 4(M) x 2(N); each wave computes a
// 16x64 patch via four v_wmma_f32_16x16x4_f32 accumulators. BM=64 BN=128 BK=32.
//
// LDS layouts:
//  As[64][36]   : 36-float row stride -> the 4-bank-group fragment pattern
//                 (lane m: banks 36m+{0..3} mod 64) covers all 64 banks once.
//  Bs[16][144]  : K-paired float2 {B[k][n],B[k+1][n]}; row stride 288 floats
//                 == 32 banks mod 64, so lanes 0-15 (row k/2) and 16-31
//                 (row k/2+1) hit complementary bank halves -> conflict-free
//                 single ds_load_b64 per B operand, no v_mov assembly.
// RELU_A applies relu to A elements at load time (fused layer-1 activation).
// ---------------------------------------------------------------------------
template <bool RELU_A>
__global__ __launch_bounds__(256) void gemm_wmma_f32(
    const float* __restrict__ A, const float* __restrict__ B,
    float* __restrict__ C, int M, int N, int K) {
    constexpr int BM = 64, BN = 128, BK = 32;
    constexpr int BNP = BN + 16;              // float2 row pad (see above)
    __shared__ float  As[BM][BK + 4];
    __shared__ float2 Bs[BK / 2][BNP];

    const int tid  = threadIdx.x;
    const int wave = tid >> 5;      // 0..7 (wave32)
    const int lane = tid & 31;
    const int wm   = wave >> 1;     // 0..3  (16-row slab)
    const int wn   = wave & 1;      // 0..1  (64-col slab)

    const int blockM = blockIdx.x * BM;
    const int blockN = blockIdx.y * BN;

    v8f acc[4] = {v8f{}, v8f{}, v8f{}, v8f{}};

    // A 16x4 fragment layout (ISA 7.12.2): lanes 0-15 -> K=k,k+1 ; lanes 16-31 -> K=k+2,k+3
    const int am   = lane & 15;
    const int aks  = (lane >> 4) * 2;         // 0 | 2
    const int bcol = wn * 64 + (lane & 15);

    for (int k0 = 0; k0 < K; k0 += BK) {
        // ---- stage A tile (64x32), coalesced float4, zero-fill OOB rows ----
        {
            int r  = tid >> 3;          // 0..31
            int c4 = (tid & 7) * 4;     // 0..28
#pragma unroll
            for (int half = 0; half < 2; ++half) {
                int row = r + half * 32;
                int gr  = blockM + row;
                float4 v = make_float4(0.f, 0.f, 0.f, 0.f);
                if (gr < M) {
                    const float* ap = A + (size_t)gr * K + k0 + c4;
                    v = *(const float4*)ap;
                    if (k0 + BK < K)  // prefetch next K-slab (global_prefetch_b8)
                        __builtin_prefetch(ap + BK, 0, 3);
                }
                if (RELU_A) {
                    v.x = fmaxf(v.x, 0.f); v.y = fmaxf(v.y, 0.f);
                    v.z = fmaxf(v.z, 0.f); v.w = fmaxf(v.w, 0.f);
                }
                *(float4*)&As[row][c4] = v;
            }
        }
        // ---- stage B tile (32x128) K-paired: Bs[k/2][n] = {B[k][n], B[k+1][n]} ----
        {
            int n      = tid & 127;     // lane-consecutive -> conflict-free b64 stores
            int k2base = tid >> 7;      // 0..1
#pragma unroll
            for (int it = 0; it < 8; ++it) {
                int k2 = k2base + it * 2;
                const float* bp = B + (size_t)(k0 + 2 * k2) * N + blockN + n;
                Bs[k2][n] = make_float2(bp[0], bp[N]);
            }
        }
        __syncthreads();

#pragma unroll
        for (int kk = 0; kk < BK; kk += 4) {
            v2f a;
            a.x = As[wm * 16 + am][kk + aks + 0];
            a.y = As[wm * 16 + am][kk + aks + 1];
            const int kidx = (kk + aks) >> 1;   // even pair index (aligned b64)
#pragma unroll
            for (int nt = 0; nt < 4; ++nt) {
                v2f b = *(const v2f*)&Bs[kidx][bcol + nt * 16];
                acc[nt] = __builtin_amdgcn_wmma_f32_16x16x4_f32(
                    false, a, false, b, (short)0, acc[nt], false, false);
            }
        }
        __syncthreads();
    }

    // C/D layout: VGPR r, lane<16 -> row r, col lane ; lane>=16 -> row r+8, col lane-16
    const int rbase = blockM + wm * 16 + ((lane >> 4) << 3);
    const int cbase = blockN + wn * 64 + (lane & 15);
#pragma unroll
    for (int r = 0; r < 8; ++r) {
        int row = rbase + r;
        if (row < M) {
#pragma unroll
            for (int nt = 0; nt < 4; ++nt)
                C[(size_t)row * N + cbase + nt * 16] = acc[nt][r];
        }
    }
}

// ---------------------------------------------------------------------------
// out[i,c] = h[i,c] * dinv[i]^2 + b[c]   (self-loop message + bias; also
// initializes the accumulation buffer so edge atomics can add on top)
// ---------------------------------------------------------------------------
__global__ void selfloop_bias_kernel(const float* __restrict__ h,
                                     const float* __restrict__ dinv,
                                     const float* __restrict__ b,
                                     float* __restrict__ out, int n, int Cc) {
    int i = blockIdx.x * blockDim.x + threadIdx.x;
    int total = n * Cc;
    if (i >= total) return;
    int row = i / Cc;
    int col = i - row * Cc;
    float dv = dinv[row];
    out[i] = h[i] * dv * dv + b[col];
}

// ---------------------------------------------------------------------------
// Per-edge gather + scaled scatter-add.  C/4 threads per edge, float4 gather
// (L2-resident: 10000 x C x 4B << 192MB), per-component global f32 atomics.
// ---------------------------------------------------------------------------
template <int Cc>
__global__ __launch_bounds__(256) void edge_aggregate_kernel(
    const float* __restrict__ h, const float* __restrict__ dinv,
    const int* __restrict__ src, const int* __restrict__ dst,
    float* __restrict__ out, int E) {
    constexpr int TPE = Cc / 4;        // threads per edge
    constexpr int EPB = 256 / TPE;     // edges per block
    int e = blockIdx.x * EPB + threadIdx.x / TPE;
    if (e >= E) return;
    int t = threadIdx.x % TPE;
    int s = src[e];
    int d = dst[e];
    float norm = dinv[s] * dinv[d];
    float4 v = *(const float4*)(h + (size_t)s * Cc + t * 4);
    float* o = out + (size_t)d * Cc + t * 4;
    atomicAdd(o + 0, v.x * norm);
    atomicAdd(o + 1, v.y * norm);
    atomicAdd(o + 2, v.z * norm);
    atomicAdd(o + 3, v.w * norm);
}

// ---------------------------------------------------------------------------
// Launch
// ---------------------------------------------------------------------------
extern "C" void kernel_launch(void* const* d_in, const int* in_sizes, int n_in,
                              void* d_out, int out_size, void* d_ws, size_t ws_size,
                              hipStream_t stream) {
    const float* x  = (const float*)d_in[0];
    const int*   ei = (const int*)d_in[1];       // int64 in ref, but JAX x64-off -> int32
    const float* W1 = (const float*)d_in[2];
    const float* b1 = (const float*)d_in[3];
    const float* W2 = (const float*)d_in[4];
    const float* b2 = (const float*)d_in[5];

    const int hid   = in_sizes[3];               // 512
    const int in_c  = in_sizes[2] / hid;         // 512
    const int out_c = in_sizes[5];               // 256
    const int Nn    = in_sizes[0] / in_c;        // 10000
    const int E     = in_sizes[1] / 2;           // 160000
    const int* src  = ei;
    const int* dst  = ei + E;

    // workspace layout (floats)
    float* deg  = (float*)d_ws;
    float* dinv = deg + ((Nn + 63) & ~63);
    float* h1   = dinv + ((Nn + 63) & ~63);
    float* a1   = h1 + (size_t)Nn * hid;         // layer-1 aggregated output
    float* h2   = a1 + (size_t)Nn * hid;         // layer-2 pre-aggregation
    float* z    = (float*)d_out;                 // layer-2 aggregated output

    // --- degrees & D^-1/2 (shared by both layers) ---
    deg_init_kernel<<<(Nn + 255) / 256, 256, 0, stream>>>(deg, Nn);
    deg_accum_kernel<<<(E + 255) / 256, 256, 0, stream>>>(dst, deg, E);
    dinv_kernel<<<(Nn + 255) / 256, 256, 0, stream>>>(deg, dinv, Nn);

    // --- layer 1: h1 = x @ W1 ---
    {
        dim3 grid((Nn + 63) / 64, hid / 128);
        gemm_wmma_f32<false><<<grid, 256, 0, stream>>>(x, W1, h1, Nn, hid, in_c);
    }
    // a1 = selfloop(h1) + b1 ; then edge scatter-add
    selfloop_bias_kernel<<<((size_t)Nn * hid + 255) / 256, 256, 0, stream>>>(h1, dinv, b1, a1, Nn, hid);
    edge_aggregate_kernel<512><<<(E * (512 / 4) + 255) / 256, 256, 0, stream>>>(h1, dinv, src, dst, a1, E);

    // --- layer 2: h2 = relu(a1) @ W2  (relu fused into A load) ---
    {
        dim3 grid((Nn + 63) / 64, out_c / 128);
        gemm_wmma_f32<true><<<grid, 256, 0, stream>>>(a1, W2, h2, Nn, out_c, hid);
    }
    selfloop_bias_kernel<<<((size_t)Nn * out_c + 255) / 256, 256, 0, stream>>>(h2, dinv, b2, z, Nn, out_c);
    edge_aggregate_kernel<256><<<(E * (256 / 4) + 255) / 256, 256, 0, stream>>>(h2, dinv, src, dst, z, E);
}